// MixHop_4501125726318
// MI455X (gfx1250) — compile-verified
//
#include <hip/hip_runtime.h>

typedef float v2f __attribute__((ext_vector_type(2)));
typedef float v8f __attribute__((ext_vector_type(8)));

#define BN_EPS 1e-5f

static __device__ __forceinline__ v8f wmma_f32(v2f a, v2f b, v8f c) {
  // V_WMMA_F32_16X16X4_F32 : D(16x16 f32) = A(16x4 f32) x B(4x16 f32) + C
  return __builtin_amdgcn_wmma_f32_16x16x4_f32(false, a, false, b, (short)0, c,
                                               false, false);
}

// ---------------------------------------------------------------------------
// Utility kernels
// ---------------------------------------------------------------------------
__global__ void fill_zero_f(float* __restrict__ p, long n) {
  long i = blockIdx.x * (long)blockDim.x + threadIdx.x;
  if (i < n) p[i] = 0.0f;
}

__global__ void fill_zero_i(int* __restrict__ p, long n) {
  long i = blockIdx.x * (long)blockDim.x + threadIdx.x;
  if (i < n) p[i] = 0;
}

// ---------------------------------------------------------------------------
// gcn_norm + CSR(dest-major) build
// ---------------------------------------------------------------------------
__global__ void deg_count(const int* __restrict__ col, int* __restrict__ cnt,
                          int E) {
  int e = blockIdx.x * blockDim.x + threadIdx.x;
  if (e < E) atomicAdd(&cnt[col[e]], 1);
}

// Single-block exclusive scan of cnt[N] -> rowptr[N+1]; also copies to cursor.
__global__ __launch_bounds__(1024) void exscan_rowptr(
    const int* __restrict__ cnt, int N, int* __restrict__ rowptr,
    int* __restrict__ cursor) {
  __shared__ int sdata[1024];
  __shared__ int s_running;
  const int t = threadIdx.x;
  if (t == 0) s_running = 0;
  __syncthreads();
  for (int base = 0; base < N; base += 1024) {
    int i = base + t;
    int v = (i < N) ? cnt[i] : 0;
    sdata[t] = v;
    __syncthreads();
    int x = v;  // inclusive Hillis-Steele scan
    for (int off = 1; off < 1024; off <<= 1) {
      int y = (t >= off) ? sdata[t - off] : 0;
      __syncthreads();
      x += y;
      sdata[t] = x;
      __syncthreads();
    }
    int run = s_running;
    if (i < N) {
      int ex = run + x - v;
      rowptr[i] = ex;
      cursor[i] = ex;
    }
    __syncthreads();  // all reads of s_running done
    if (t == 1023) s_running = run + x;
    __syncthreads();
  }
  if (t == 0) rowptr[N] = s_running;
}

__global__ void make_dinv(const int* __restrict__ cnt, float* __restrict__ dinv,
                          int n) {
  int i = blockIdx.x * blockDim.x + threadIdx.x;
  if (i < n) {
    int c = cnt[i];
    dinv[i] = c > 0 ? rsqrtf((float)c) : 0.0f;
  }
}

__global__ void csr_fill(const int* __restrict__ row, const int* __restrict__ col,
                         const float* __restrict__ dinv, int* __restrict__ cursor,
                         int* __restrict__ csr_src, float* __restrict__ csr_w,
                         int E) {
  int e = blockIdx.x * blockDim.x + threadIdx.x;
  if (e < E) {
    int r = row[e], c = col[e];
    int pos = atomicAdd(&cursor[c], 1);
    csr_src[pos] = r;
    csr_w[pos] = dinv[r] * dinv[c];
  }
}

// Gather SpMM: out[v, c] = sum_{j in in-edges(v)} w[j] * in[src[j], c]
// 64 threads (= feature columns) per destination node, 4 nodes per block.
// j0/j1/j are wave-uniform (a wave32 covers half the columns of one node);
// readfirstlane makes that explicit so src[j]/w[j] can go scalar.
__global__ __launch_bounds__(256) void spmm_csr(
    const int* __restrict__ rowptr, const int* __restrict__ src,
    const float* __restrict__ w, const float* __restrict__ xin, int ldin,
    float* __restrict__ xout, int ldout, int N) {
  int v = blockIdx.x * 4 + (threadIdx.x >> 6);
  int c = threadIdx.x & 63;
  if (v >= N) return;
  int j0 = __builtin_amdgcn_readfirstlane(rowptr[v]);
  int j1 = __builtin_amdgcn_readfirstlane(rowptr[v + 1]);
  float acc = 0.0f;
  for (int j = j0; j < j1; ++j) {
    int s = __builtin_amdgcn_readfirstlane(src[j]);
    float wj = w[j];
    acc += wj * xin[(size_t)s * ldin + c];
  }
  xout[(size_t)v * ldout + c] = acc;
}

// ---------------------------------------------------------------------------
// BatchNorm (axis 0), 192 channels
// ---------------------------------------------------------------------------
__global__ __launch_bounds__(192) void bn_partial(const float* __restrict__ h,
                                                  int N,
                                                  float* __restrict__ sums) {
  const int c = threadIdx.x;
  long r0 = blockIdx.x * 256L;
  float s = 0.0f, s2 = 0.0f;
  for (int i = 0; i < 256; ++i) {
    long r = r0 + i;
    if (r < N) {
      float v = h[(size_t)r * 192 + c];
      s += v;
      s2 += v * v;
    }
  }
  atomicAdd(&sums[c], s);
  atomicAdd(&sums[192 + c], s2);
}

__global__ void bn_finalize(const float* __restrict__ sums,
                            const float* __restrict__ gamma,
                            const float* __restrict__ beta,
                            float* __restrict__ scale,
                            float* __restrict__ shift, float invN) {
  int c = threadIdx.x;
  float mu = sums[c] * invN;
  float var = sums[192 + c] * invN - mu * mu;
  float sc = gamma[c] * rsqrtf(var + BN_EPS);
  scale[c] = sc;
  shift[c] = beta[c] - mu * sc;
}

__global__ __launch_bounds__(192) void bn_apply_relu(
    float* __restrict__ h, int N, const float* __restrict__ scale,
    const float* __restrict__ shift) {
  const int c = threadIdx.x;
  const float sc = scale[c];
  const float sh = shift[c];
  long r0 = blockIdx.x * 4L;
  for (int i = 0; i < 4; ++i) {
    long r = r0 + i;
    if (r < N) {
      size_t o = (size_t)r * 192 + c;
      h[o] = fmaxf(h[o] * sc + sh, 0.0f);
    }
  }
}

// ---------------------------------------------------------------------------
// Fused multi-hop WMMA GEMM: for h in [0,NH): Y_h = X @ W_h + b_h
//   X: N x K (ld ldx), W: contiguous [NH][K][64] row-major, Y_h: N x 64.
//
// W is staged once per block into LDS, k-interleaved: sW[h][k/4][n][4] holds
// W[k..k+3][n] in 4 consecutive dwords. A B-fragment (W[kk][n], W[kk+1][n])
// is then one aligned ds_load_b64 directly into the WMMA operand pair —
// no register shuffles — and the 32 lanes of a fragment read hit 32 distinct
// LDS banks (lane l, half hf -> dwords 4l+2hf, 4l+2hf+1): conflict-free.
//
// A frag (16x4 f32): lanes 0-15 hold K=k,k+1 of row M=lane; lanes 16-31 hold
// K=k+2,k+3. C/D frag: vgpr r -> M=r / M=r+8, N=lane&15. (ISA 7.12.2)
// ---------------------------------------------------------------------------
template <int NH>
__global__ __launch_bounds__(256) void gemm_wmma(
    const float* __restrict__ X, int ldx, int K, int N,
    const float* __restrict__ W, const float* __restrict__ B, int bstride,
    float* __restrict__ Y0, int ldy0, float* __restrict__ Y1,
    float* __restrict__ Y2) {
  extern __shared__ float sW[];  // NH*K*64 floats, k-interleaved

  // cooperative stage: sW[h*K*64 + (k>>2)*256 + n*4 + (k&3)] = W[h][k][n]
  const int perh = K * 64;
#pragma unroll
  for (int h = 0; h < NH; ++h) {
    for (int i = threadIdx.x; i < perh; i += 256) {
      int k = i >> 6;
      int n = i & 63;
      sW[h * perh + ((k >> 2) << 8) + (n << 2) + (k & 3)] = W[h * perh + i];
    }
  }
  __syncthreads();

  const int lane = threadIdx.x & 31;
  const int wave = threadIdx.x >> 5;
  const int tile = blockIdx.x * 8 + wave;
  const int nTiles = (N + 15) >> 4;
  if (tile >= nTiles) return;  // wave-uniform exit after barrier

  const int half = lane >> 4;  // 0: K sub-cols {0,1}; 1: {2,3}
  const int l = lane & 15;

  int arow = tile * 16 + l;
  if (arow >= N) arow = N - 1;  // clamp (tail tile); stores masked below
  const float* Xrow = X + (size_t)arow * ldx + half * 2;

  v8f acc[NH][4];
#pragma unroll
  for (int h = 0; h < NH; ++h)
#pragma unroll
    for (int t = 0; t < 4; ++t)
      acc[h][t] = (v8f){0.f, 0.f, 0.f, 0.f, 0.f, 0.f, 0.f, 0.f};

  // per-lane base into the interleaved LDS image (dwords): l*4 + half*2
  const int lbase = (l << 2) + (half << 1);

  for (int k = 0; k < K; k += 4) {
    v2f a;
    a.x = Xrow[k];
    a.y = Xrow[k + 1];
    const int kbase = (k >> 2) << 8;  // (k/4)*256
#pragma unroll
    for (int h = 0; h < NH; ++h) {
      const float* Wh = sW + h * perh + kbase + lbase;
#pragma unroll
      for (int t = 0; t < 4; ++t) {
        v2f b = *(const v2f*)(Wh + (t << 6));  // one ds_load_b64
        acc[h][t] = wmma_f32(a, b, acc[h][t]);
      }
    }
  }

  const int row0 = tile * 16 + half * 8;
  float* Ys[3] = {Y0, Y1, Y2};
  int ldys[3] = {ldy0, 64, 64};
#pragma unroll
  for (int h = 0; h < NH; ++h) {
    float* __restrict__ Yh = Ys[h];
    const int ldy = ldys[h];
#pragma unroll
    for (int t = 0; t < 4; ++t) {
      const float bias = B[h * bstride + t * 16 + l];
#pragma unroll
      for (int r = 0; r < 8; ++r) {
        int m = row0 + r;
        if (m < N) Yh[(size_t)m * ldy + t * 16 + l] = acc[h][t][r] + bias;
      }
    }
  }
}

// ---------------------------------------------------------------------------
// Host launcher
// ---------------------------------------------------------------------------
extern "C" void kernel_launch(void* const* d_in, const int* in_sizes, int n_in,
                              void* d_out, int out_size, void* d_ws,
                              size_t ws_size, hipStream_t stream) {
  const float* x = (const float*)d_in[0];
  const int* ei = (const int*)d_in[1];
  // d_in[2] = n (device scalar, derived from sizes instead)
  const float* l0w = (const float*)d_in[3];
  const float* l0b = (const float*)d_in[4];
  const float* l1w = (const float*)d_in[5];
  const float* l1b = (const float*)d_in[6];
  const float* gamma = (const float*)d_in[7];
  const float* beta = (const float*)d_in[8];
  const float* fpw = (const float*)d_in[9];
  const float* fpb = (const float*)d_in[10];

  const int IN = 128;
  const int N = in_sizes[0] / IN;
  const int E = in_sizes[1] / 2;
  const int* row = ei;
  const int* col = ei + E;

  // workspace layout
  char* wsb = (char*)d_ws;
  int* cnt = (int*)wsb;          wsb += sizeof(int) * (size_t)N;
  int* rowptr = (int*)wsb;       wsb += sizeof(int) * (size_t)(N + 1);
  int* cursor = (int*)wsb;       wsb += sizeof(int) * (size_t)N;
  int* csr_src = (int*)wsb;      wsb += sizeof(int) * (size_t)E;
  float* csr_w = (float*)wsb;    wsb += sizeof(float) * (size_t)E;
  float* dinv = (float*)wsb;     wsb += sizeof(float) * (size_t)N;
  float* bufA = (float*)wsb;     wsb += sizeof(float) * (size_t)N * 64;
  float* bufB = (float*)wsb;     wsb += sizeof(float) * (size_t)N * 64;
  float* h = (float*)wsb;        wsb += sizeof(float) * (size_t)N * 192;
  float* h2 = (float*)wsb;       wsb += sizeof(float) * (size_t)N * 192;
  float* bns = (float*)wsb;      wsb += sizeof(float) * 384;
  float* bnscale = (float*)wsb;  wsb += sizeof(float) * 192;
  float* bnshift = (float*)wsb;  wsb += sizeof(float) * 192;

  const unsigned eb = (unsigned)((E + 255) / 256);
  const unsigned nb = (unsigned)((N + 255) / 256);
  const int nTiles = (N + 15) / 16;
  const unsigned gemmBlocks = (unsigned)((nTiles + 7) / 8);
  const unsigned spmmBlocks = (unsigned)((N + 3) / 4);

  // ---- gcn_norm + dest-major CSR build (graph reused by all 6 SpMMs) ----
  fill_zero_i<<<nb, 256, 0, stream>>>(cnt, N);
  deg_count<<<eb, 256, 0, stream>>>(col, cnt, E);
  exscan_rowptr<<<1, 1024, 0, stream>>>(cnt, N, rowptr, cursor);
  make_dinv<<<nb, 256, 0, stream>>>(cnt, dinv, N);
  csr_fill<<<eb, 256, 0, stream>>>(row, col, dinv, cursor, csr_src, csr_w, E);

  // ---- layer 0: MixHop(x; W0) -> h (N x 192) ----
  gemm_wmma<3><<<gemmBlocks, 256, (size_t)3 * IN * 64 * 4, stream>>>(
      x, IN, IN, N, l0w, l0b, 64, h, 192, bufA, bufB);
  spmm_csr<<<spmmBlocks, 256, 0, stream>>>(rowptr, csr_src, csr_w, bufA, 64,
                                           h + 64, 192, N);
  spmm_csr<<<spmmBlocks, 256, 0, stream>>>(rowptr, csr_src, csr_w, bufB, 64,
                                           bufA, 64, N);
  spmm_csr<<<spmmBlocks, 256, 0, stream>>>(rowptr, csr_src, csr_w, bufA, 64,
                                           h + 128, 192, N);

  // ---- batchnorm + relu on h ----
  fill_zero_f<<<2, 256, 0, stream>>>(bns, 384);
  bn_partial<<<nb, 192, 0, stream>>>(h, N, bns);
  bn_finalize<<<1, 192, 0, stream>>>(bns, gamma, beta, bnscale, bnshift,
                                     1.0f / (float)N);
  bn_apply_relu<<<(N + 3) / 4, 192, 0, stream>>>(h, N, bnscale, bnshift);

  // ---- layer 1: MixHop(h; W1) -> h2 (N x 192) ----
  gemm_wmma<3><<<gemmBlocks, 256, (size_t)3 * 192 * 64 * 4, stream>>>(
      h, 192, 192, N, l1w, l1b, 64, h2, 192, bufA, bufB);
  spmm_csr<<<spmmBlocks, 256, 0, stream>>>(rowptr, csr_src, csr_w, bufA, 64,
                                           h2 + 64, 192, N);
  spmm_csr<<<spmmBlocks, 256, 0, stream>>>(rowptr, csr_src, csr_w, bufB, 64,
                                           bufA, 64, N);
  spmm_csr<<<spmmBlocks, 256, 0, stream>>>(rowptr, csr_src, csr_w, bufA, 64,
                                           h2 + 128, 192, N);

  // ---- final projection: out = h2 @ fp_w + fp_b ----
  gemm_wmma<1><<<gemmBlocks, 256, (size_t)192 * 64 * 4, stream>>>(
      h2, 192, 192, N, fpw, fpb, 0, (float*)d_out, 64, nullptr, nullptr);
}